// SleepClassifier_32598801776875
// MI455X (gfx1250) — compile-verified
//
#include <hip/hip_runtime.h>

typedef float v2f __attribute__((ext_vector_type(2)));
typedef float v4f __attribute__((ext_vector_type(4)));
typedef float v8f __attribute__((ext_vector_type(8)));

__device__ __forceinline__ float eluf(float x) {
    return x > 0.f ? x : (__expf(x) - 1.f);
}

__device__ __forceinline__ v8f wmma_k4(v2f a, v2f b, v8f c) {
    // D = A(16x4 f32) * B(4x16 f32) + C(16x16 f32)
    return __builtin_amdgcn_wmma_f32_16x16x4_f32(
        false, a, false, b, (short)0, c, false, false);
}

// One wave (32 threads) per block; each wave processes 16 independent
// 8-timestep sequences through the 2-layer BiLSTM aggregator.
// Layer 1 (H=10): z = [x_t | h | 1] @ [W;U;b] via V_WMMA_F32_16X16X4_F32,
//   operands kept in an LDS tile AB[t][row][16] -> unconditional ds_load_b64.
// Layer 2 (H=1): input projection P[t] = [hs_f|hs_b] @ v + bias via WMMA
//   (no recurrence), then the rank-1 h*r recurrence scalar over 32 lanes.
template <int CIN>
__global__ __launch_bounds__(32) void lstm_agg_kernel(
    const float* __restrict__ in,   // [Nchunks, 8, CIN]
    float* __restrict__ out,        // [Nchunks]
    const float* __restrict__ wfp, const float* __restrict__ ufp, const float* __restrict__ bfp,
    const float* __restrict__ wbp, const float* __restrict__ ubp, const float* __restrict__ bbp,
    const float* __restrict__ vfp, const float* __restrict__ rfp, const float* __restrict__ cfp,
    const float* __restrict__ vbp, const float* __restrict__ rbp, const float* __restrict__ cbp)
{
    constexpr int KTOT   = CIN + 10;       // x cols + h cols
    constexpr int KQ     = KTOT + 1;       // + bias column (A == 1)
    constexpr int KSTEPS = (KQ + 3) / 4;   // WMMA k-steps of 4

    __shared__ float AB[8][16][16];        // [t][row][k]: x | h | 1 | 0-pad
    __shared__ float zb[48][16];           // gate pre-acts, col-major [n][m]
    __shared__ float hs[8][16][20];        // [t][row][fwd 0..9 | bwd 10..19]
    __shared__ float Pb[2][8][4][16];      // layer-2 projection [dir][t][gate][row]

    const int lane  = threadIdx.x;         // 0..31 (wave32)
    const int row   = lane & 15;
    const int khalf = lane >> 4;           // 0 -> K=k0,k0+1 ; 1 -> K=k0+2,k0+3
    const int col   = row;                 // N column owned by this lane
    const int mbase = khalf * 8;           // C tile: VGPR r -> M = mbase + r
    const long long seq_base = (long long)blockIdx.x * 16;

    // ---- zero AB (vectorized), set bias column, stage x ----
    for (int i = lane; i < 8 * 16 * 16 / 4; i += 32)
        ((v4f*)&AB[0][0][0])[i] = (v4f){0.f, 0.f, 0.f, 0.f};
    __syncthreads();
    for (int i = lane; i < 8 * 16; i += 32) AB[i >> 4][i & 15][KTOT] = 1.f;
    for (int i = lane; i < 16 * 8 * CIN; i += 32) {
        int r = i / (8 * CIN), rem = i % (8 * CIN);
        AB[rem / CIN][r][rem % CIN] = in[(seq_base + r) * (8 * CIN) + rem];
    }
    __syncthreads();  // single-wave workgroup -> S_NOP

    // ================= Layer 1: BiLSTM H=10 =================
    for (int dir = 0; dir < 2; ++dir) {
        const float* w = dir ? wbp : wfp;  // [CIN, 40]
        const float* u = dir ? ubp : ufp;  // [10, 40]
        const float* b = dir ? bbp : bfp;  // [40]

        // B fragments: rows k<CIN from w, CIN..KTOT-1 from u, KTOT = bias
        v2f bfrag[3][KSTEPS];
#pragma unroll
        for (int nt = 0; nt < 3; ++nt) {
            int n = nt * 16 + col;
#pragma unroll
            for (int kk = 0; kk < KSTEPS; ++kk) {
                int k0 = kk * 4 + khalf * 2, k1 = k0 + 1;
                float b0 = 0.f, b1 = 0.f;
                if (n < 40) {
                    b0 = (k0 < CIN) ? w[k0 * 40 + n]
                       : (k0 < KTOT) ? u[(k0 - CIN) * 40 + n]
                       : (k0 == KTOT) ? b[n] : 0.f;
                    b1 = (k1 < CIN) ? w[k1 * 40 + n]
                       : (k1 < KTOT) ? u[(k1 - CIN) * 40 + n]
                       : (k1 == KTOT) ? b[n] : 0.f;
                }
                v2f bv; bv.x = b0; bv.y = b1;
                bfrag[nt][kk] = bv;
            }
        }

        // cell state: lane owns pairs p = lane + 32q -> (row=p/10, hh=p%10)
        float cst[5];
#pragma unroll
        for (int q = 0; q < 5; ++q) cst[q] = 0.f;
        // zero h columns of the starting timestep
        {
            int t0 = dir ? 7 : 0;
            for (int i = lane; i < 160; i += 32) AB[t0][i / 10][CIN + i % 10] = 0.f;
        }
        __syncthreads();

        for (int step = 0; step < 8; ++step) {
            const int t = dir ? (7 - step) : step;

            v8f c0 = {}, c1 = {}, c2 = {};
            const float* arow = &AB[t][row][khalf * 2];
#pragma unroll
            for (int kk = 0; kk < KSTEPS; ++kk) {
                v2f a = *(const v2f*)(arow + kk * 4);   // ds_load_b64, no branches
                c0 = wmma_k4(a, bfrag[0][kk], c0);
                c1 = wmma_k4(a, bfrag[1][kk], c1);
                c2 = wmma_k4(a, bfrag[2][kk], c2);
            }

            // spill z col-major: two ds_store_b128 per tile
            *(v4f*)&zb[col][mbase]          = __builtin_shufflevector(c0, c0, 0, 1, 2, 3);
            *(v4f*)&zb[col][mbase + 4]      = __builtin_shufflevector(c0, c0, 4, 5, 6, 7);
            *(v4f*)&zb[16 + col][mbase]     = __builtin_shufflevector(c1, c1, 0, 1, 2, 3);
            *(v4f*)&zb[16 + col][mbase + 4] = __builtin_shufflevector(c1, c1, 4, 5, 6, 7);
            *(v4f*)&zb[32 + col][mbase]     = __builtin_shufflevector(c2, c2, 0, 1, 2, 3);
            *(v4f*)&zb[32 + col][mbase + 4] = __builtin_shufflevector(c2, c2, 4, 5, 6, 7);
            __syncthreads();

            const int tn = dir ? (t - 1) : (t + 1);
            const bool wr_next = ((unsigned)tn < 8u);   // wave-uniform
#pragma unroll
            for (int q = 0; q < 5; ++q) {
                int p = lane + 32 * q, rr = p / 10, hh = p % 10;
                float iv = eluf(zb[hh][rr]);
                float fv = eluf(zb[10 + hh][rr]);
                float gv = eluf(zb[20 + hh][rr]);       // act = elu
                float ov = eluf(zb[30 + hh][rr]);
                float cv = fv * cst[q] + iv * gv;
                cst[q] = cv;
                float hv = ov * eluf(cv);               // h = elu(o) * elu(c)
                hs[t][rr][dir * 10 + hh] = hv;
                if (wr_next) AB[tn][rr][CIN + hh] = hv;
            }
            __syncthreads();
        }
    }

    // ================= Layer 2: projection via WMMA =================
    for (int dir = 0; dir < 2; ++dir) {
        const float* v  = dir ? vbp : vfp;  // [20, 4]
        const float* cb = dir ? cbp : cfp;  // [4]
        v2f bf2[5];
#pragma unroll
        for (int kk = 0; kk < 5; ++kk) {
            int k0 = kk * 4 + khalf * 2;
            float b0 = 0.f, b1 = 0.f;
            if (col < 4) { b0 = v[k0 * 4 + col]; b1 = v[(k0 + 1) * 4 + col]; }
            v2f bv; bv.x = b0; bv.y = b1;
            bf2[kk] = bv;
        }
        float bias2 = (col < 4) ? cb[col] : 0.f;

        for (int t = 0; t < 8; ++t) {
            v8f p;
#pragma unroll
            for (int r = 0; r < 8; ++r) p[r] = bias2;
            const float* srow = &hs[t][row][khalf * 2];
#pragma unroll
            for (int kk = 0; kk < 5; ++kk) {
                v2f a = *(const v2f*)(srow + kk * 4);   // ds_load_b64
                p = wmma_k4(a, bf2[kk], p);
            }
            if (col < 4) {
                *(v4f*)&Pb[dir][t][col][mbase]     = __builtin_shufflevector(p, p, 0, 1, 2, 3);
                *(v4f*)&Pb[dir][t][col][mbase + 4] = __builtin_shufflevector(p, p, 4, 5, 6, 7);
            }
        }
    }
    __syncthreads();

    // ---- rank-1 recurrence, fused across lanes: lane = dir*16 + row ----
    {
        const int dir = khalf;
        const float* rw = dir ? rbp : rfp;  // [4]
        float r0 = rw[0], r1 = rw[1], r2 = rw[2], r3 = rw[3];
        float h = 0.f, c = 0.f;
        for (int step = 0; step < 8; ++step) {
            int t = dir ? (7 - step) : step;
            float z0 = Pb[dir][t][0][row] + h * r0;
            float z1 = Pb[dir][t][1][row] + h * r1;
            float z2 = Pb[dir][t][2][row] + h * r2;   // act = identity
            float z3 = Pb[dir][t][3][row] + h * r3;
            float iv = eluf(z0), fv = eluf(z1), ov = eluf(z3);
            c = fv * c + iv * z2;
            h = ov * c;                               // h = elu(o) * c
        }
        (&zb[0][0])[lane] = h;
    }
    __syncthreads();
    if (lane < 16) {
        float m = 0.5f * ((&zb[0][0])[lane] + (&zb[0][0])[lane + 16]);
        out[seq_base + lane] = 1.f / (1.f + __expf(-m));
    }
}

extern "C" void kernel_launch(void* const* d_in, const int* in_sizes, int n_in,
                              void* d_out, int out_size, void* d_ws, size_t ws_size,
                              hipStream_t stream) {
    (void)in_sizes; (void)n_in; (void)out_size; (void)ws_size;

    const float* x   = (const float*)d_in[0];
    const float* w0f = (const float*)d_in[1];
    const float* u0f = (const float*)d_in[2];
    const float* b0f = (const float*)d_in[3];
    const float* w0b = (const float*)d_in[4];
    const float* u0b = (const float*)d_in[5];
    const float* b0b = (const float*)d_in[6];
    const float* w1f = (const float*)d_in[7];   // [3,1,40]
    const float* u1f = (const float*)d_in[8];   // [3,10,40]
    const float* b1f = (const float*)d_in[9];   // [3,40]
    const float* w1b = (const float*)d_in[10];
    const float* u1b = (const float*)d_in[11];
    const float* b1b = (const float*)d_in[12];
    const float* v_f = (const float*)d_in[13];  // [4,20,4]
    const float* r_f = (const float*)d_in[14];  // [4,1,4]
    const float* c_f = (const float*)d_in[15];  // [4,4]
    const float* v_b = (const float*)d_in[16];
    const float* r_b = (const float*)d_in[17];
    const float* c_b = (const float*)d_in[18];

    float* ws = (float*)d_ws;
    float* o0 = ws;             // 524288 floats
    float* o1 = o0 + 524288;    // 65536 floats
    float* o2 = o1 + 65536;     // 8192 floats
    float* out = (float*)d_out; // 1024 floats

    // Stage 0: Cin=5, 524288 chunks -> 32768 waves
    lstm_agg_kernel<5><<<32768, 32, 0, stream>>>(
        x, o0,
        w0f, u0f, b0f, w0b, u0b, b0b,
        v_f + 0 * 80, r_f + 0 * 4, c_f + 0 * 4,
        v_b + 0 * 80, r_b + 0 * 4, c_b + 0 * 4);

    // Stage 1: Cin=1, 65536 chunks
    lstm_agg_kernel<1><<<4096, 32, 0, stream>>>(
        o0, o1,
        w1f + 0 * 40, u1f + 0 * 400, b1f + 0 * 40,
        w1b + 0 * 40, u1b + 0 * 400, b1b + 0 * 40,
        v_f + 1 * 80, r_f + 1 * 4, c_f + 1 * 4,
        v_b + 1 * 80, r_b + 1 * 4, c_b + 1 * 4);

    // Stage 2: Cin=1, 8192 chunks
    lstm_agg_kernel<1><<<512, 32, 0, stream>>>(
        o1, o2,
        w1f + 1 * 40, u1f + 1 * 400, b1f + 1 * 40,
        w1b + 1 * 40, u1b + 1 * 400, b1b + 1 * 40,
        v_f + 2 * 80, r_f + 2 * 4, c_f + 2 * 4,
        v_b + 2 * 80, r_b + 2 * 4, c_b + 2 * 4);

    // Stage 3: Cin=1, 1024 chunks -> final [16,64,1] output
    lstm_agg_kernel<1><<<64, 32, 0, stream>>>(
        o2, out,
        w1f + 2 * 40, u1f + 2 * 400, b1f + 2 * 40,
        w1b + 2 * 40, u1b + 2 * 400, b1b + 2 * 40,
        v_f + 3 * 80, r_f + 3 * 4, c_f + 3 * 4,
        v_b + 3 * 80, r_b + 3 * 4, c_b + 3 * 4);
}